// CountPrediction_37151467110675
// MI455X (gfx1250) — compile-verified
//
#include <hip/hip_runtime.h>

// CountPrediction: elementwise over 1M cells, single gene column (g=17).
// HBM-bound (~100+ MB traffic dominated by the 800B-strided t[:,g] gather).
// No matrix ops in the math -> no WMMA; we lean on gfx1250 cache controls:
// NT-hinted loads for the uncacheable 800MB t matrix, NT streaming stores,
// and b128-coalesced access for everything else.

typedef float v4f __attribute__((ext_vector_type(4)));

namespace {

constexpr int CELLS = 1000000;
constexpr int GENES = 200;
constexpr int CPT   = 4;                   // cells per thread
constexpr int NTHREADS = CELLS / CPT;      // 250000 (CELLS divisible by 4)

struct US { float u; float s; };

__device__ __forceinline__ float fast_rcp(float x) {
    return __builtin_amdgcn_rcpf(x);
}

__device__ __forceinline__ US cell_math(float alpha, float gamma, float beta,
                                        float tg, float kg, float dg,
                                        float t0g, float u0g) {
    // reference's beta/gamma fixups
    beta  = (beta  == 0.0f)  ? beta + 0.75f : beta;
    beta  = (beta  == gamma) ? beta + 0.75f : beta;
    gamma = (gamma == 0.0f)  ? gamma + 0.75f : gamma;

    const float S   = fast_rcp(1.0f + __expf(-(kg * (tg - t0g - dg))));  // sigmoid
    const float eb  = __expf(-beta  * tg);
    const float eg  = __expf(-gamma * tg);
    const float eb0 = __expf(-beta  * (tg - t0g));
    const float eg0 = __expf(-gamma * (tg - t0g));

    const float ab    = alpha * fast_rcp(beta);
    const float ag    = alpha * fast_rcp(gamma);
    const float rgb   = fast_rcp(gamma - beta);
    const float oneMS = 1.0f - S;

    US r;
    r.u = ab * (1.0f - eb) * oneMS + ab * S + (u0g * eb0 - ab) * S;
    r.s = (ag * (1.0f - eg) + alpha * rgb * (eg - eb)) * oneMS
        + ag * S
        + beta * u0g * rgb * (eg0 - eb0) * S;
    return r;
}

__global__ __launch_bounds__(256)
void count_pred_kernel(const float* __restrict__ out3,   // CELLS x 3
                       const float* __restrict__ kv,     // GENES
                       const float* __restrict__ dv,     // GENES
                       const float* __restrict__ t0v,    // GENES
                       const float* __restrict__ u0v,    // GENES
                       const float* __restrict__ tmat,   // CELLS x GENES
                       const int*   __restrict__ gidx,   // scalar
                       float*       __restrict__ res)    // [tilde_u | tilde_s], 2*CELLS
{
    const int tid = blockIdx.x * blockDim.x + threadIdx.x;
    if (tid >= NTHREADS) return;

    const int   g   = *gidx;
    const float kg  = kv[g];
    const float dg  = dv[g];
    const float t0g = t0v[g];
    const float u0g = u0v[g];

    // out rows for 4 consecutive cells: 12 contiguous floats -> 3x b128
    const v4f* o4 = reinterpret_cast<const v4f*>(out3) + (size_t)tid * 3;
    const v4f p0 = o4[0];
    const v4f p1 = o4[1];
    const v4f p2 = o4[2];

    // strided column gather from the 800MB t matrix: NT hint so each
    // wasted cacheline doesn't evict useful L2 contents.
    const float* tcol = tmat + (size_t)tid * CPT * GENES + g;
    const float tg0 = __builtin_nontemporal_load(tcol + 0 * GENES);
    const float tg1 = __builtin_nontemporal_load(tcol + 1 * GENES);
    const float tg2 = __builtin_nontemporal_load(tcol + 2 * GENES);
    const float tg3 = __builtin_nontemporal_load(tcol + 3 * GENES);

    const US r0 = cell_math(p0.x, p0.y, p0.z, tg0, kg, dg, t0g, u0g);
    const US r1 = cell_math(p0.w, p1.x, p1.y, tg1, kg, dg, t0g, u0g);
    const US r2 = cell_math(p1.z, p1.w, p2.x, tg2, kg, dg, t0g, u0g);
    const US r3 = cell_math(p2.y, p2.z, p2.w, tg3, kg, dg, t0g, u0g);

    v4f u, s;
    u.x = r0.u; u.y = r1.u; u.z = r2.u; u.w = r3.u;
    s.x = r0.s; s.y = r1.s; s.z = r2.s; s.w = r3.s;

    // streaming (write-once) outputs: NT b128 stores
    v4f* up = reinterpret_cast<v4f*>(res) + tid;
    v4f* sp = reinterpret_cast<v4f*>(res + CELLS) + tid;
    __builtin_nontemporal_store(u, up);
    __builtin_nontemporal_store(s, sp);
}

} // anonymous namespace

extern "C" void kernel_launch(void* const* d_in, const int* in_sizes, int n_in,
                              void* d_out, int out_size, void* d_ws, size_t ws_size,
                              hipStream_t stream) {
    const float* out3 = (const float*)d_in[0];
    const float* kv   = (const float*)d_in[1];
    const float* dv   = (const float*)d_in[2];
    const float* t0v  = (const float*)d_in[3];
    const float* u0v  = (const float*)d_in[4];
    const float* tmat = (const float*)d_in[5];
    const int*   gi   = (const int*)d_in[6];
    float*       res  = (float*)d_out;

    const int threads = 256;                                // 8 wave32 per block
    const int blocks  = (NTHREADS + threads - 1) / threads; // 977
    count_pred_kernel<<<blocks, threads, 0, stream>>>(out3, kv, dv, t0v, u0v,
                                                      tmat, gi, res);
}